// DHGCF1_11269994184845
// MI455X (gfx1250) — compile-verified
//
#include <hip/hip_runtime.h>
#include <math.h>
#include <stdint.h>

#define N_NODES 200000
#define DIM     128
#define NNZ_E   6400000
#define EPSF    1e-12f

typedef __attribute__((ext_vector_type(2))) float v2f;
typedef __attribute__((ext_vector_type(8))) float v8f;

// ---------------- zero fill (float4 grid-stride) ----------------
__global__ void dhgcf_zero_f4(float4* __restrict__ p, int n4) {
    int i = blockIdx.x * blockDim.x + threadIdx.x;
    int stride = gridDim.x * blockDim.x;
    float4 z; z.x = 0.f; z.y = 0.f; z.z = 0.f; z.w = 0.f;
    for (; i < n4; i += stride) p[i] = z;
}

// ---------------- SpMM: one wave32 per edge ----------------
// y[rows[e], :] += vals[e] * x[cols[e], :]   (D=128, each lane owns 4 floats)
__global__ __launch_bounds__(256)
void dhgcf_spmm(const int* __restrict__ rows, const int* __restrict__ cols,
                const float* __restrict__ vals, const float* __restrict__ x,
                float* __restrict__ y) {
    int gwave = (blockIdx.x * blockDim.x + threadIdx.x) >> 5;
    int lane  = threadIdx.x & 31;
    if (gwave >= NNZ_E) return;
    int r = rows[gwave];
    int c = cols[gwave];
    float v = vals[gwave];
    const float4* xp = (const float4*)(x + (size_t)c * DIM);
    float4 xv = xp[lane];                      // coalesced 512B row gather (L2-resident)
    float* yp = y + (size_t)r * DIM + lane * 4;
    __hip_atomic_fetch_add(yp + 0, v * xv.x, __ATOMIC_RELAXED, __HIP_MEMORY_SCOPE_AGENT);
    __hip_atomic_fetch_add(yp + 1, v * xv.y, __ATOMIC_RELAXED, __HIP_MEMORY_SCOPE_AGENT);
    __hip_atomic_fetch_add(yp + 2, v * xv.z, __ATOMIC_RELAXED, __HIP_MEMORY_SCOPE_AGENT);
    __hip_atomic_fetch_add(yp + 3, v * xv.w, __ATOMIC_RELAXED, __HIP_MEMORY_SCOPE_AGENT);
}

// ---------------- GEMM(128) + bias + row L2 normalize ----------------
// Block: 256 threads = 8 waves. Block handles rows [r0, r0+16).
// Wave w computes the 16x16 tile at columns [16w, 16w+16) using
// 32x V_WMMA_F32_16X16X4_F32 over K=128. A-stripe staged to LDS via
// GLOBAL_LOAD_ASYNC_TO_LDS_B128 (ASYNCcnt path, no VGPR round-trip).
__global__ __launch_bounds__(256)
void dhgcf_gemm_bias_norm(const float* __restrict__ side,
                          const float* __restrict__ W,
                          const float* __restrict__ bias,
                          float* __restrict__ out) {
    __shared__ __align__(16) float As[16 * 132];   // padded stride: no bank conflicts
    __shared__ float rowsq[16];

    const int r0 = blockIdx.x * 16;
    const int t  = threadIdx.x;

    // async-stage 16x128 A-stripe into LDS (each thread: 2 x 16B direct-to-LDS)
    {
        int f = t * 4;
#pragma unroll
        for (int rep = 0; rep < 2; ++rep) {
            int r = f >> 7, c = f & 127;
            // low 32 bits of the flat shared-pointer == LDS byte offset
            uint32_t lds_off = (uint32_t)(uintptr_t)(&As[r * 132 + c]);
            uint64_t gaddr   = (uint64_t)(uintptr_t)(side + (size_t)(r0 + r) * DIM + c);
            asm volatile("global_load_async_to_lds_b128 %0, %1, off"
                         :: "v"(lds_off), "v"(gaddr) : "memory");
            f += 1024;
        }
    }
    if (t < 16) rowsq[t] = 0.f;
    asm volatile("s_wait_asynccnt 0" ::: "memory");
    __syncthreads();

    const int wave = t >> 5;       // 0..7 -> column tile
    const int lane = t & 31;
    const int n0   = wave * 16;
    const int li   = lane & 15;    // A: row m; B/C/D: col n
    const int kh   = lane >> 4;    // lane half

    v8f acc = {};
#pragma unroll 8
    for (int kb = 0; kb < 32; ++kb) {
        int k0 = kb * 4 + kh * 2;
        v2f a, b;
        a.x = As[li * 132 + k0];
        a.y = As[li * 132 + k0 + 1];
        b.x = W[(size_t)(k0)     * DIM + n0 + li];
        b.y = W[(size_t)(k0 + 1) * DIM + n0 + li];
        acc = __builtin_amdgcn_wmma_f32_16x16x4_f32(
            /*neg_a=*/false, a, /*neg_b=*/false, b,
            /*c_mod=*/(short)0, acc, /*reuse_a=*/false, /*reuse_b=*/false);
    }

    // bias (broadcast over rows; column = n0+li for every acc element)
    float bv = bias[n0 + li];
#pragma unroll
    for (int i = 0; i < 8; ++i) acc[i] += bv;

    // per-row sum of squares: xor masks < 16 never cross a 16-lane half,
    // so full-width shuffles need no clamping logic.
#pragma unroll
    for (int i = 0; i < 8; ++i) {
        float s = acc[i] * acc[i];
        s += __shfl_xor(s, 1, 32);
        s += __shfl_xor(s, 2, 32);
        s += __shfl_xor(s, 4, 32);
        s += __shfl_xor(s, 8, 32);
        if (li == 0)
            __hip_atomic_fetch_add(&rowsq[i + 8 * kh], s,
                                   __ATOMIC_RELAXED, __HIP_MEMORY_SCOPE_WORKGROUP);
    }
    __syncthreads();

    // scale by 1/max(||row||, eps) and store (coalesced 16-float runs)
#pragma unroll
    for (int i = 0; i < 8; ++i) {
        int row = i + 8 * kh;
        float scale = 1.0f / fmaxf(sqrtf(rowsq[row]), EPSF);
        out[(size_t)(r0 + row) * DIM + n0 + li] = acc[i] * scale;
    }
}

extern "C" void kernel_launch(void* const* d_in, const int* in_sizes, int n_in,
                              void* d_out, int out_size, void* d_ws, size_t ws_size,
                              hipStream_t stream) {
    (void)in_sizes; (void)n_in; (void)out_size; (void)ws_size;
    const float* fts  = (const float*)d_in[0];
    const float* vals = (const float*)d_in[1];
    const float* W0   = (const float*)d_in[2];
    const float* b0   = (const float*)d_in[3];
    const float* W1   = (const float*)d_in[4];
    const float* b1   = (const float*)d_in[5];
    const int*   rows = (const int*)d_in[6];
    const int*   cols = (const int*)d_in[7];
    float* out  = (float*)d_out;
    float* side = (float*)d_ws;                  // N*D floats = 102.4 MB scratch

    const int n4 = N_NODES * DIM / 4;
    const int spmm_blocks = NNZ_E / 8;           // 8 waves (edges) per 256-thr block
    const int gemm_blocks = N_NODES / 16;        // 12500

    // ---- layer 0: side = A @ fts ; out = norm(side@W0 + b0) ----
    dhgcf_zero_f4<<<4096, 256, 0, stream>>>((float4*)side, n4);
    dhgcf_spmm<<<spmm_blocks, 256, 0, stream>>>(rows, cols, vals, fts, side);
    dhgcf_gemm_bias_norm<<<gemm_blocks, 256, 0, stream>>>(side, W0, b0, out);

    // ---- layer 1: side = A @ out ; out = norm(side@W1 + b1) ----
    dhgcf_zero_f4<<<4096, 256, 0, stream>>>((float4*)side, n4);
    dhgcf_spmm<<<spmm_blocks, 256, 0, stream>>>(rows, cols, vals, out, side);
    dhgcf_gemm_bias_norm<<<gemm_blocks, 256, 0, stream>>>(side, W1, b1, out);
}